// TGaBP_57286273794621
// MI455X (gfx1250) — compile-verified
//
#include <hip/hip_runtime.h>

typedef float v2f __attribute__((ext_vector_type(2)));
typedef float v8f __attribute__((ext_vector_type(8)));

#define M_DIM 128
#define N_DIM 128
#define NITER 16

__device__ __forceinline__ float fast_rcp(float x) {
    return __builtin_amdgcn_rcpf(x);   // v_rcp_f32 (~1 ulp), no div-fixup chain
}
__device__ __forceinline__ float fast_sigmoid(float x) {
    return fast_rcp(1.0f + __expf(-x));
}
__device__ __forceinline__ float fast_tanh(float x) {
    return 1.0f - 2.0f * fast_rcp(__expf(2.0f * x) + 1.0f);
}

// One workgroup (512 threads = 16 wave32) per received symbol k.
// Matrix layout: (m,n) tile grid 8x8 of 16x16 WMMA-C-layout tiles.
// wave w: tile-row r = w>>1, column half h = w&1 -> tiles (r, 4h+c), c=0..3.
// lane l: hi = l>>4, li = l&15; element (m = 16r + j + 8hi, n = 64h+16c+li),
// j = 0..7 -> per-lane state = 32 floats per matrix (pure VGPR at the
// 4-waves/SIMD VGPR budget declared by __launch_bounds__(512)).
// State kept: SR (soft replica), DL = ER - SR^2 (variance), UU, VV.
__global__ void __launch_bounds__(512)
tgabp_kernel(const float* __restrict__ y,
             const float* __restrict__ H,
             const float* __restrict__ N0,
             const float* __restrict__ eta,
             const float* __restrict__ mu,
             const float* __restrict__ lay,
             const float* __restrict__ norm,
             const float* __restrict__ esmax,
             float* __restrict__ out) {
    __shared__ float sgL[NITER * N_DIM];  // sigmoid(4*eta) gates, 8KB
    __shared__ float PA[8][N_DIM];        // per-tile-row colsum partials of Rec
    __shared__ float PB[8][N_DIM];        // per-tile-row colsum partials of elem
    __shared__ float CA[N_DIM];           // colsum(Rec)
    __shared__ float CB[N_DIM];           // colsum(elem)
    __shared__ float RUU[M_DIM];          // rowsum(uu)
    __shared__ float RVV[M_DIM];          // rowsum(vv)
    __shared__ float RU[M_DIM];           // rowsum(u)  (last iter only)
    __shared__ float RV[M_DIM];           // rowsum(v)  (last iter only)

    const int k    = blockIdx.x;
    const int tid  = threadIdx.x;
    const int wv   = tid >> 5;          // 0..15
    const int lane = tid & 31;
    const int hi   = lane >> 4;
    const int li   = lane & 15;
    const int r    = wv >> 1;           // tile row 0..7
    const int h2   = wv & 1;            // column half 0..1
    const int mBase = 16 * r + 8 * hi;  // first of this lane's 8 rows
    const int nBase = 64 * h2 + li;     // column for c=0 (c adds 16c)

    const float n0h     = 0.5f * N0[0];
    const float nrm     = norm[0];
    const float inv_nrm = fast_rcp(nrm);
    const float esm_h   = 0.5f * esmax[0];
    const float lay0 = lay[0], lay1 = lay[1], lay2 = lay[2];

    // Precompute damping gates sigmoid(4*eta[n, it]) once (eta is (N, NITER)).
    for (int t = tid; t < NITER * N_DIM; t += 512) {
        int n  = t & (N_DIM - 1);
        int it = t >> 7;
        sgL[t] = fast_sigmoid(4.0f * eta[n * NITER + it]);
    }

    // Register-resident H^T tile and y slice. H is (N, M) row-major.
    float hreg[4][8], y4[4];
#pragma unroll
    for (int c = 0; c < 4; ++c) {
        const int n = nBase + 16 * c;
        y4[c] = y[k * N_DIM + n];
#pragma unroll
        for (int j = 0; j < 8; ++j)
            hreg[c][j] = H[n * M_DIM + (mBase + j)];
    }

    float SR[4][8], DL[4][8], UU[4][8], VV[4][8];
#pragma unroll
    for (int c = 0; c < 4; ++c)
#pragma unroll
        for (int j = 0; j < 8; ++j) {
            SR[c][j] = 0.0f; DL[c][j] = 0.5f;  // ER=0.5, SR=0 -> delta=0.5
            UU[c][j] = 0.0f; VV[c][j] = 0.0f;
        }

    __syncthreads();

    for (int it = 0; it < NITER; ++it) {
        const float mu_i = mu[it];
        const bool  last = (it == NITER - 1);

        // Zero row accumulators (covered by the phase-A barrier below).
        if (tid < M_DIM) {
            RUU[tid] = 0.0f; RVV[tid] = 0.0f;
            if (last) { RU[tid] = 0.0f; RV[tid] = 0.0f; }
        }

        // ---- Phase A: per-wave column partials of Rec = Ht*SR, elem = HH*delta
#pragma unroll
        for (int c = 0; c < 4; ++c) {
            float a = 0.0f, b = 0.0f;
#pragma unroll
            for (int j = 0; j < 8; ++j) {
                float h = hreg[c][j];
                a += h * SR[c][j];
                b += h * h * DL[c][j];
            }
            a += __shfl_xor(a, 16, 32);  // fold hi=0 / hi=1 row halves
            b += __shfl_xor(b, 16, 32);
            if (hi == 0) {
                PA[r][nBase + 16 * c] = a;
                PB[r][nBase + 16 * c] = b;
            }
        }
        __syncthreads();

        // ---- WMMA colsum: reduce 8x128 partials with ones(16x4) x P(4x16)
        // chunks, accumulated over K=8. D rows replicate the column sums, so
        // every lane's acc[0] holds colsum of its column. Waves 0-7 -> CA,
        // waves 8-15 -> CB (both matrices reduced in parallel on the XDL pipe,
        // co-executing with the other waves' VALU work). All 16 waves run this
        // uniformly -> EXEC is all ones as WMMA requires.
        {
            const float* P    = (wv < 8) ? &PA[0][0] : &PB[0][0];
            float*       Cdst = (wv < 8) ? CA : CB;
            const int    t8   = wv & 7;
            v2f a; a.x = 1.0f; a.y = 1.0f;   // all-ones A (16x4)
            v8f acc = {};
#pragma unroll
            for (int qk = 0; qk < 2; ++qk) {
                // B fragment (4x16): VGPR0 rows K=4qk+2hi, VGPR1 rows K=4qk+1+2hi
                v2f b;
                b.x = P[(4 * qk + 2 * hi)     * N_DIM + t8 * 16 + li];
                b.y = P[(4 * qk + 1 + 2 * hi) * N_DIM + t8 * 16 + li];
                acc = __builtin_amdgcn_wmma_f32_16x16x4_f32(
                    false, a, false, b, (short)0, acc, false, false);
            }
            if (hi == 0) Cdst[t8 * 16 + li] = acc[0];
        }
        __syncthreads();

        // ---- Phase B: psi, y_tilde, u, v, damped uu/vv, row sums.
        // j-outer so only 4 scalar partials are live at a time. Damping in
        // lerp form uu + sg*(u - uu) avoids keeping (1-sg) live.
        float ymc[4], csB[4], sg4[4];
        const float* sgRow = &sgL[it * N_DIM];
#pragma unroll
        for (int c = 0; c < 4; ++c) {
            const int n = nBase + 16 * c;
            ymc[c] = y4[c] - CA[n];
            csB[c] = CB[n];
            sg4[c] = sgRow[n];
        }
#pragma unroll
        for (int j = 0; j < 8; ++j) {
            float suu = 0.0f, svv = 0.0f, su = 0.0f, sv = 0.0f;
#pragma unroll
            for (int c = 0; c < 4; ++c) {
                float h   = hreg[c][j], sr = SR[c][j];
                float hh  = h * h;
                float psi = csB[c] - hh * DL[c][j] + n0h;  // leave-one-out over M
                float yt  = ymc[c] + h * sr;               // leave-one-out over M
                float ip  = fast_rcp(psi);
                float u   = h * yt * ip;
                float v   = hh * ip;
                float nuu = UU[c][j] + sg4[c] * (u - UU[c][j]);
                float nvv = VV[c][j] + sg4[c] * (v - VV[c][j]);
                UU[c][j] = nuu; VV[c][j] = nvv;
                suu += nuu; svv += nvv;
                if (last) { su += u; sv += v; }
            }
#pragma unroll
            for (int msk = 1; msk < 16; msk <<= 1) {  // reduce 16-lane halves
                suu += __shfl_xor(suu, msk, 32);
                svv += __shfl_xor(svv, msk, 32);
                if (last) {
                    su += __shfl_xor(su, msk, 32);
                    sv += __shfl_xor(sv, msk, 32);
                }
            }
            if (li == 0) {
                const int m = mBase + j;
                atomicAdd(&RUU[m], suu);
                atomicAdd(&RVV[m], svv);
                if (last) { atomicAdd(&RU[m], su); atomicAdd(&RV[m], sv); }
            }
        }
        __syncthreads();

        // ---- Phase C: extrinsic means + tanh soft replica; store SR and
        // delta = ER - SR^2 directly.
        const float g = mu_i * inv_nrm;
#pragma unroll
        for (int j = 0; j < 8; ++j) {
            const int   m   = mBase + j;
            const float ruu = RUU[m];
            const float rvv = RVV[m];
#pragma unroll
            for (int c = 0; c < 4; ++c) {
                float s     = ruu - UU[c][j];   // leave-one-out over N
                float omega = rvv - VV[c][j];
                float gamma = s * fast_rcp(omega);
                float t0 = fast_tanh(g * (gamma - lay0));
                float t1 = fast_tanh(g * (gamma - lay1));
                float t2 = fast_tanh(g * (gamma - lay2));
                float srn = nrm * (t0 + t1 + t2);
                float ern = 2.0f * nrm * (t0 * lay0 + t1 * lay1 + t2 * lay2) + esm_h;
                SR[c][j] = srn;
                DL[c][j] = ern - srn * srn;
            }
        }
        __syncthreads();
    }

    // ---- Final soft decision from posterior means gp = sum(u)/sum(v)
    if (tid < M_DIM) {
        float gp = RU[tid] * fast_rcp(RV[tid]);
        float g  = mu[NITER - 1] * inv_nrm;
        float t0 = fast_tanh(g * (gp - lay0));
        float t1 = fast_tanh(g * (gp - lay1));
        float t2 = fast_tanh(g * (gp - lay2));
        out[k * M_DIM + tid] = nrm * (t0 + t1 + t2);
    }
}

extern "C" void kernel_launch(void* const* d_in, const int* in_sizes, int n_in,
                              void* d_out, int out_size, void* d_ws, size_t ws_size,
                              hipStream_t stream) {
    // setup_inputs order: x, y, H, N0, eta, mu, lay, norm, Esmax
    const float* y     = (const float*)d_in[1];
    const float* H     = (const float*)d_in[2];
    const float* N0    = (const float*)d_in[3];
    const float* eta   = (const float*)d_in[4];
    const float* mu    = (const float*)d_in[5];
    const float* lay   = (const float*)d_in[6];
    const float* norm  = (const float*)d_in[7];
    const float* esmax = (const float*)d_in[8];
    float* out = (float*)d_out;

    const int K = in_sizes[1] / N_DIM;  // number of symbols
    tgabp_kernel<<<K, 512, 0, stream>>>(y, H, N0, eta, mu, lay, norm, esmax, out);
}